// SelectiveScan_24043226923320
// MI455X (gfx1250) — compile-verified
//
#include <hip/hip_runtime.h>

// Selective scan (Mamba-like), 3-pass chunked formulation for MI455X (gfx1250).
//
// Roofline: A+B = 512MB dominate; pure HBM-bandwidth problem (~23us ideal at
// 23.3 TB/s). Sequential L-scan limits parallelism, so we chunk L into NC=16
// pieces: pass1 (local decay product P + local end-state E), pass2 (tiny carry
// scan over chunks), pass3 (rescan with correct initial state, emit y).
// 2048 independent waves in passes 1/3; wide b128 loads + unroll-4 keep ~4
// timesteps (~9KB) in flight per wave -> ~18MB device-wide, enough to cover
// HBM latency at full bandwidth. A/B streamed non-temporal so the 192MB L2
// holds x/C/carry tensors instead of a dead 512MB stream.
//
// WMMA: y_t[d] = sum_n h[d,n]*C_t[n] via chained V_WMMA_F32_16X16X4_F32.
// Per-lane n-mapping n(j,K) = 2j + (K&1) + 8*(K>>1): lane l owns the
// contiguous n-range 8*(l>>4)..8*(l>>4)+7 -> all state I/O is two b128 ops.

typedef float v2f __attribute__((ext_vector_type(2)));
typedef float v4f __attribute__((ext_vector_type(4)));
typedef float v8f __attribute__((ext_vector_type(8)));

#define BATCH  2
#define LSEQ   2048
#define DINNER 1024
#define NSTATE 16
#define NCHUNK 16
#define CKLEN  (LSEQ / NCHUNK)          // 128
#define DBLK   (DINNER / 16)            // 64
#define NWAVES (BATCH * DBLK * NCHUNK)  // 2048

#define LO2(v) __builtin_shufflevector((v), (v), 0, 1)
#define HI2(v) __builtin_shufflevector((v), (v), 2, 3)

static __device__ __forceinline__ v4f nt_load4(const float* p) {
  return __builtin_nontemporal_load((const v4f*)p);
}

// ---------------------------------------------------------------------------
// Pass 1: per chunk, compute P = cumprod(a) and E = local scan end (h0 = 0).
// One wave handles (b, 16-d block, chunk). Lane l: d = l&15, owns the 8
// contiguous n values 8*(l>>4) .. 8*(l>>4)+7.
// ---------------------------------------------------------------------------
__global__ void ss_pass1(const float* __restrict__ A, const float* __restrict__ B,
                         const float* __restrict__ X,
                         float* __restrict__ P, float* __restrict__ E) {
  const int lane = threadIdx.x & 31;
  const int w    = blockIdx.x * (blockDim.x >> 5) + (threadIdx.x >> 5);
  const int dblk = w % DBLK;
  const int c    = (w / DBLK) % NCHUNK;
  const int b    = w / (DBLK * NCHUNK);
  const int hi   = lane >> 4;
  const int dl   = lane & 15;
  const int d0   = dblk * 16;
  const int nofs = 8 * hi;

  v4f h[2], p[2];
#pragma unroll
  for (int k = 0; k < 2; ++k) {
    h[k] = (v4f){0.f, 0.f, 0.f, 0.f};
    p[k] = (v4f){1.f, 1.f, 1.f, 1.f};
  }

  size_t rowbase = ((size_t)(b * LSEQ + c * CKLEN) * DINNER + d0 + dl) * NSTATE + nofs;
  size_t xbase   = (size_t)(b * LSEQ + c * CKLEN) * DINNER + d0 + dl;

#pragma unroll 4
  for (int i = 0; i < CKLEN; ++i) {
    float xv = X[xbase];
    v4f av0 = nt_load4(A + rowbase);
    v4f av1 = nt_load4(A + rowbase + 4);
    v4f bv0 = nt_load4(B + rowbase);
    v4f bv1 = nt_load4(B + rowbase + 4);
    h[0] = av0 * h[0] + bv0 * xv;
    h[1] = av1 * h[1] + bv1 * xv;
    p[0] = p[0] * av0;
    p[1] = p[1] * av1;
    rowbase += (size_t)DINNER * NSTATE;
    xbase   += DINNER;
  }

  size_t obase = ((size_t)((b * NCHUNK + c) * DINNER) + d0 + dl) * NSTATE + nofs;
  *(v4f*)(P + obase)     = p[0];
  *(v4f*)(P + obase + 4) = p[1];
  *(v4f*)(E + obase)     = h[0];
  *(v4f*)(E + obase + 4) = h[1];
}

// ---------------------------------------------------------------------------
// Pass 2: carry recurrence across chunks. H_init[0]=0; H_init[c+1] = P_c*H + E_c.
// Parallel over all BATCH*DINNER*NSTATE = 32768 states; 16 sequential steps.
// ---------------------------------------------------------------------------
__global__ void ss_pass2(const float* __restrict__ P, const float* __restrict__ E,
                         float* __restrict__ Hinit) {
  const int t     = blockIdx.x * blockDim.x + threadIdx.x;
  const int plane = DINNER * NSTATE;
  const int b     = t / plane;
  const int r     = t % plane;
  float H = 0.f;
  for (int c = 0; c < NCHUNK; ++c) {
    size_t g = (size_t)(b * NCHUNK + c) * plane + r;
    Hinit[g] = H;
    H = P[g] * H + E[g];
  }
}

// ---------------------------------------------------------------------------
// Pass 3: rescan each chunk from its correct H_init; contract with C_t via
// chained v_wmma_f32_16x16x4_f32; add x*D; store y.
// Tile j covers n(j,K) = 2j + (K&1) + 8*(K>>1); B-tile/A-tile operands are
// v2f slices of the two contiguous v4f registers per lane.
// ---------------------------------------------------------------------------
__global__ void ss_pass3(const float* __restrict__ A, const float* __restrict__ B,
                         const float* __restrict__ C, const float* __restrict__ X,
                         const float* __restrict__ Dv, const float* __restrict__ Hinit,
                         float* __restrict__ Y) {
  const int lane = threadIdx.x & 31;
  const int w    = blockIdx.x * (blockDim.x >> 5) + (threadIdx.x >> 5);
  const int dblk = w % DBLK;
  const int c    = (w / DBLK) % NCHUNK;
  const int b    = w / (DBLK * NCHUNK);
  const int hi   = lane >> 4;
  const int dl   = lane & 15;
  const int d0   = dblk * 16;
  const int nofs = 8 * hi;

  // Initial state for this chunk (same layout pass1 wrote).
  size_t hbase = ((size_t)((b * NCHUNK + c) * DINNER) + d0 + dl) * NSTATE + nofs;
  v4f h0 = *(const v4f*)(Hinit + hbase);
  v4f h1 = *(const v4f*)(Hinit + hbase + 4);

  const float dcoef = Dv[d0 + dl];

  size_t rowbase = ((size_t)(b * LSEQ + c * CKLEN) * DINNER + d0 + dl) * NSTATE + nofs;
  size_t xbase   = (size_t)(b * LSEQ + c * CKLEN) * DINNER + d0 + dl;
  size_t cbase   = (size_t)(b * LSEQ + c * CKLEN) * NSTATE + nofs;
  size_t ybase   = (size_t)(b * LSEQ + c * CKLEN) * DINNER + d0 + lane;

#pragma unroll 4
  for (int i = 0; i < CKLEN; ++i) {
    float xv = X[xbase];

    // A-operand source: C_t broadcast over rows M; per lane C[t*16 + 8hi + 0..7].
    v4f cv0 = *(const v4f*)(C + cbase);
    v4f cv1 = *(const v4f*)(C + cbase + 4);

    // Elementwise state update: h = a*h + b*x (layout-agnostic).
    v4f av0 = nt_load4(A + rowbase);
    v4f av1 = nt_load4(A + rowbase + 4);
    v4f bv0 = nt_load4(B + rowbase);
    v4f bv1 = nt_load4(B + rowbase + 4);
    h0 = av0 * h0 + bv0 * xv;
    h1 = av1 * h1 + bv1 * xv;

    // y[d0+N] = sum_n C[n]*h[N,n] : 4 chained 16x16x4 f32 WMMAs over v2f tiles.
    v8f acc = {0.f, 0.f, 0.f, 0.f, 0.f, 0.f, 0.f, 0.f};
    acc = __builtin_amdgcn_wmma_f32_16x16x4_f32(false, LO2(cv0), false, LO2(h0),
                                                (short)0, acc, false, false);
    acc = __builtin_amdgcn_wmma_f32_16x16x4_f32(false, HI2(cv0), false, HI2(h0),
                                                (short)0, acc, false, false);
    acc = __builtin_amdgcn_wmma_f32_16x16x4_f32(false, LO2(cv1), false, LO2(h1),
                                                (short)0, acc, false, false);
    acc = __builtin_amdgcn_wmma_f32_16x16x4_f32(false, HI2(cv1), false, HI2(h1),
                                                (short)0, acc, false, false);

    // Every row M of D holds y; lanes 0-15 carry (M=0, N=lane).
    if (lane < 16) {
      Y[ybase] = acc[0] + xv * dcoef;
    }

    rowbase += (size_t)DINNER * NSTATE;
    xbase   += DINNER;
    cbase   += NSTATE;
    ybase   += DINNER;
  }
}

extern "C" void kernel_launch(void* const* d_in, const int* in_sizes, int n_in,
                              void* d_out, int out_size, void* d_ws, size_t ws_size,
                              hipStream_t stream) {
  const float* A  = (const float*)d_in[0];
  const float* B  = (const float*)d_in[1];
  const float* C  = (const float*)d_in[2];
  const float* X  = (const float*)d_in[3];
  const float* Dv = (const float*)d_in[4];
  float* Y = (float*)d_out;

  // Workspace: P, E, Hinit, each (BATCH, NCHUNK, DINNER, NSTATE) f32 = 2MB.
  const size_t planeBytes = (size_t)BATCH * NCHUNK * DINNER * NSTATE * sizeof(float);
  float* P  = (float*)d_ws;
  float* E  = (float*)((char*)d_ws + planeBytes);
  float* Hi = (float*)((char*)d_ws + 2 * planeBytes);

  dim3 blk(256);  // 8 waves per block (wave32)
  ss_pass1<<<NWAVES / 8, blk, 0, stream>>>(A, B, X, P, E);
  ss_pass2<<<(BATCH * DINNER * NSTATE) / 256, blk, 0, stream>>>(P, E, Hi);
  ss_pass3<<<NWAVES / 8, blk, 0, stream>>>(A, B, C, X, Dv, Hi, Y);
}